// Net_45140106281501
// MI455X (gfx1250) — compile-verified
//
#include <hip/hip_runtime.h>
#include <hip/hip_bf16.h>
#include <math.h>

// ---------------------------------------------------------------------------
// GCN (3 layers, H=64) + classifier + entropy-weighted ragged pooling.
// fp32 end-to-end; matmuls use V_WMMA_F32_16X16X4_F32 (CDNA5 wave32 WMMA).
// ---------------------------------------------------------------------------

typedef __attribute__((ext_vector_type(2))) float v2f;
typedef __attribute__((ext_vector_type(8))) float v8f;

#define HDIM 64
#define CDIM 10
#define EPS  1e-5f

// ---------------------------------------------------------------------------
// Edge scatter-add: out[dst[e], :] += hin[src[e], :]
// 16 threads per edge, each thread moves one float4 chunk (coalesced gather,
// 4x global_atomic_add_f32 scatter). L2 (192MB) fully holds the 25.6MB dest.
// ---------------------------------------------------------------------------
__global__ void gcn_scatter_add(const float* __restrict__ hin,
                                const int* __restrict__ src,
                                const int* __restrict__ dst,
                                float* __restrict__ out, int nedges) {
  int t  = blockIdx.x * blockDim.x + threadIdx.x;
  int e  = t >> 4;        // edge index
  int f4 = t & 15;        // float4 chunk within the 64-float row
  if (e >= nedges) return;
  const float4* s = (const float4*)(hin + (size_t)src[e] * HDIM) + f4;
  float4 v = *s;
  float* d = out + (size_t)dst[e] * HDIM + f4 * 4;
  atomicAdd(d + 0, v.x);
  atomicAdd(d + 1, v.y);
  atomicAdd(d + 2, v.z);
  atomicAdd(d + 3, v.w);
}

// ---------------------------------------------------------------------------
// Fused: Hout = ELU( BN( M @ W + bias ) ), M is [nrows,64], W is [64,64].
// One wave32 computes a 16x64 output slab via 16 k-blocks x 4 n-tiles of
// V_WMMA_F32_16X16X4_F32. In-place safe: each wave reads only its own 16 rows
// and writes them only after all K accumulation is done.
//
// fp32 WMMA fragment layouts (ISA 7.12.2):
//   A 16x4:  lanes 0-15 -> M=lane, {K=0,K=1}; lanes 16-31 -> M=lane-16, {K=2,K=3}
//   B 4x16:  VGPR0: lanes0-15 row K=0, lanes16-31 row K=2; VGPR1: K=1 / K=3
//   C 16x16: VGPR r: lanes0-15 (M=r, N=lane), lanes16-31 (M=r+8, N=lane-16)
// ---------------------------------------------------------------------------
__global__ void gcn_gemm_bn_elu(float* __restrict__ HM,   // in/out [nrows,64]
                                const float* __restrict__ W,      // [64,64]
                                const float* __restrict__ bias,   // [64]
                                const float* __restrict__ gamma,  // [64]
                                const float* __restrict__ beta,   // [64]
                                const float* __restrict__ mean,   // [64]
                                const float* __restrict__ var,    // [64]
                                int nrows) {
  int wave = (blockIdx.x * blockDim.x + threadIdx.x) >> 5;
  int lane = threadIdx.x & 31;
  int row0 = wave * 16;
  if (row0 >= nrows) return;                 // uniform across the wave
  int hi = lane >> 4;                        // 0 or 1 (half-wave)
  int lo = lane & 15;

  if (row0 + 16 <= nrows) {
    v8f c0 = {}, c1 = {}, c2 = {}, c3 = {};
    for (int kb = 0; kb < 16; ++kb) {
      int k0 = kb * 4 + 2 * hi;              // this lane's K pair
      const float* arow = HM + (size_t)(row0 + lo) * HDIM + k0;
      v2f a;
      a.x = arow[0];
      a.y = arow[1];
      const float* w0 = W + (size_t)k0 * HDIM;        // row K=k0
      const float* w1 = W + (size_t)(k0 + 1) * HDIM;  // row K=k0+1
      v2f b0, b1, b2, b3;
      b0.x = w0[lo];      b0.y = w1[lo];
      b1.x = w0[16 + lo]; b1.y = w1[16 + lo];
      b2.x = w0[32 + lo]; b2.y = w1[32 + lo];
      b3.x = w0[48 + lo]; b3.y = w1[48 + lo];
      c0 = __builtin_amdgcn_wmma_f32_16x16x4_f32(false, a, false, b0, (short)0, c0, false, false);
      c1 = __builtin_amdgcn_wmma_f32_16x16x4_f32(false, a, false, b1, (short)0, c1, false, false);
      c2 = __builtin_amdgcn_wmma_f32_16x16x4_f32(false, a, false, b2, (short)0, c2, false, false);
      c3 = __builtin_amdgcn_wmma_f32_16x16x4_f32(false, a, false, b3, (short)0, c3, false, false);
    }
    // Epilogue: bias + BatchNorm(eval) + ELU, then store.
#pragma unroll
    for (int nt = 0; nt < 4; ++nt) {
      int col = nt * 16 + lo;
      float bb = bias[col];
      float mu = mean[col];
      float sc = gamma[col] * rsqrtf(var[col] + EPS);
      float be = beta[col];
      v8f c = (nt == 0) ? c0 : (nt == 1) ? c1 : (nt == 2) ? c2 : c3;
#pragma unroll
      for (int r = 0; r < 8; ++r) {
        int row = row0 + r + 8 * hi;
        float y = (c[r] + bb - mu) * sc + be;
        y = (y > 0.0f) ? y : (__expf(y) - 1.0f);
        HM[(size_t)row * HDIM + (nt * 16 + lo)] = y;
      }
    }
  } else {
    // Scalar tail (not taken for N=100000, kept for robustness).
    for (int r = row0; r < nrows; ++r) {
      for (int col = lane; col < HDIM; col += 32) {
        float acc = bias[col];
        for (int k = 0; k < HDIM; ++k) acc += HM[(size_t)r * HDIM + k] * W[k * HDIM + col];
        float y = (acc - mean[col]) * (gamma[col] * rsqrtf(var[col] + EPS)) + beta[col];
        y = (y > 0.0f) ? y : (__expf(y) - 1.0f);
        HM[(size_t)r * HDIM + col] = y;
      }
    }
  }
}

// ---------------------------------------------------------------------------
// Per-node: logits = h @ Wc + bc ; log_softmax ; entropy ; track max entropy.
// Wc (64x10) cached in LDS; 10-wide register accumulator per node.
// Entropy >= 0 so int-bits atomicMax is order-preserving.
// ---------------------------------------------------------------------------
__global__ void gcn_classify_entropy(const float* __restrict__ h,
                                     const float* __restrict__ Wc,
                                     const float* __restrict__ bc,
                                     float* __restrict__ hent,
                                     int* __restrict__ maxbits, int n) {
  __shared__ float sW[HDIM * CDIM];
  __shared__ float sb[CDIM];
  for (int i = threadIdx.x; i < HDIM * CDIM; i += blockDim.x) sW[i] = Wc[i];
  if (threadIdx.x < CDIM) sb[threadIdx.x] = bc[threadIdx.x];
  __syncthreads();
  int i = blockIdx.x * blockDim.x + threadIdx.x;
  if (i >= n) return;
  float acc[CDIM];
#pragma unroll
  for (int c = 0; c < CDIM; ++c) acc[c] = sb[c];
  const float* hr = h + (size_t)i * HDIM;
  for (int k = 0; k < HDIM; ++k) {
    float hv = hr[k];
#pragma unroll
    for (int c = 0; c < CDIM; ++c) acc[c] += hv * sW[k * CDIM + c];
  }
  float mx = acc[0];
#pragma unroll
  for (int c = 1; c < CDIM; ++c) mx = fmaxf(mx, acc[c]);
  float se = 0.0f;
#pragma unroll
  for (int c = 0; c < CDIM; ++c) { acc[c] -= mx; se += __expf(acc[c]); }
  float lz = __logf(se);
  float inv = 1.0f / se;
  float H = 0.0f;
#pragma unroll
  for (int c = 0; c < CDIM; ++c) {
    float p = __expf(acc[c]) * inv;
    H -= p * (acc[c] - lz);
  }
  hent[i] = H;
  atomicMax(maxbits, __float_as_int(H));
}

// ---------------------------------------------------------------------------
// pooled[g] += lam[i] * h[i],  lam = 1 - Hent/maxHent.  64 threads per node.
// ---------------------------------------------------------------------------
__global__ void gcn_lam_pool(const float* __restrict__ h,
                             const float* __restrict__ hent,
                             const int* __restrict__ maxbits,
                             const int* __restrict__ n2g,
                             float* __restrict__ pooled, int n) {
  int t = blockIdx.x * blockDim.x + threadIdx.x;
  int i = t >> 6;
  int f = t & 63;
  if (i >= n) return;
  float mh = __int_as_float(*maxbits);
  float lam = 1.0f - hent[i] / mh;
  atomicAdd(&pooled[(size_t)n2g[i] * HDIM + f], lam * h[(size_t)i * HDIM + f]);
}

// ---------------------------------------------------------------------------
// out[g,c] = bc[c] + sum_k pooled[g,k] * Wc[k,c]    (tiny: G*C = 5000)
// ---------------------------------------------------------------------------
__global__ void gcn_final_classify(const float* __restrict__ pooled,
                                   const float* __restrict__ Wc,
                                   const float* __restrict__ bc,
                                   float* __restrict__ out, int gc) {
  int t = blockIdx.x * blockDim.x + threadIdx.x;
  if (t >= gc) return;
  int g = t / CDIM;
  int c = t % CDIM;
  float acc = bc[c];
  const float* pr = pooled + (size_t)g * HDIM;
#pragma unroll 8
  for (int k = 0; k < HDIM; ++k) acc += pr[k] * Wc[k * CDIM + c];
  out[t] = acc;
}

// ---------------------------------------------------------------------------
// Host-side orchestration (graph-capture safe: only async ops on `stream`).
// ---------------------------------------------------------------------------
extern "C" void kernel_launch(void* const* d_in, const int* in_sizes, int n_in,
                              void* d_out, int out_size, void* d_ws, size_t ws_size,
                              hipStream_t stream) {
  const float* feat   = (const float*)d_in[0];   // [N,64]
  const int*   src    = (const int*)d_in[1];     // [E]
  const int*   dst    = (const int*)d_in[2];     // [E]
  const int*   n2g    = (const int*)d_in[3];     // [N]
  const float* Ws     = (const float*)d_in[4];   // [3,64,64]
  const float* bs     = (const float*)d_in[5];   // [3,64]
  const float* gammas = (const float*)d_in[6];   // [3,64]
  const float* betas  = (const float*)d_in[7];   // [3,64]
  const float* means  = (const float*)d_in[8];   // [3,64]
  const float* vars_  = (const float*)d_in[9];   // [3,64]
  const float* Wc     = (const float*)d_in[10];  // [64,10]
  const float* bc     = (const float*)d_in[11];  // [10]
  float* out = (float*)d_out;                    // [G,10]

  const int N = in_sizes[0] / HDIM;
  const int E = in_sizes[1];
  const int G = out_size / CDIM;

  // Workspace layout (floats): bufA[N*64] | bufB[N*64] | hent[N] | maxbits[1] | pooled[G*64]
  float* bufA    = (float*)d_ws;
  float* bufB    = bufA + (size_t)N * HDIM;
  float* hent    = bufB + (size_t)N * HDIM;
  int*   maxbits = (int*)(hent + N);
  float* pooled  = (float*)(maxbits + 1);

  const int L = 3;
  const size_t rowBytes = (size_t)N * HDIM * sizeof(float);

  dim3 blkScatter(256), grdScatter((unsigned)(((size_t)E * 16 + 255) / 256));
  int nWaves = (N + 15) / 16;
  dim3 blkGemm(128), grdGemm((unsigned)((nWaves + 3) / 4));

  float* cur = nullptr;
  for (int l = 0; l < L; ++l) {
    const float* hin = (l == 0) ? feat : cur;
    float* m = (l & 1) ? bufB : bufA;          // l0->A, l1->B, l2->A
    hipMemsetAsync(m, 0, rowBytes, stream);
    gcn_scatter_add<<<grdScatter, blkScatter, 0, stream>>>(hin, src, dst, m, E);
    gcn_gemm_bn_elu<<<grdGemm, blkGemm, 0, stream>>>(
        m, Ws + (size_t)l * HDIM * HDIM, bs + l * HDIM,
        gammas + l * HDIM, betas + l * HDIM, means + l * HDIM, vars_ + l * HDIM, N);
    cur = m;
  }

  hipMemsetAsync(maxbits, 0, sizeof(int), stream);
  hipMemsetAsync(pooled, 0, (size_t)G * HDIM * sizeof(float), stream);

  dim3 blkEnt(256), grdEnt((N + 255) / 256);
  gcn_classify_entropy<<<grdEnt, blkEnt, 0, stream>>>(cur, Wc, bc, hent, maxbits, N);

  dim3 blkPool(256), grdPool((unsigned)(((size_t)N * HDIM + 255) / 256));
  gcn_lam_pool<<<grdPool, blkPool, 0, stream>>>(cur, hent, maxbits, n2g, pooled, N);

  dim3 blkFin(256), grdFin((G * CDIM + 255) / 256);
  gcn_final_classify<<<grdFin, blkFin, 0, stream>>>(pooled, Wc, bc, out, G * CDIM);
}